// MultiHeadSelfAttention_71622874628520
// MI455X (gfx1250) — compile-verified
//
#include <hip/hip_runtime.h>
#include <hip/hip_bf16.h>

typedef __attribute__((ext_vector_type(2))) float v2f;
typedef __attribute__((ext_vector_type(8))) float v8f;

#define S_LEN   2048
#define D_MODEL 1024
#define NHEAD   16
#define DK      64
#define BATCH   2
#define M_TOT   (BATCH * S_LEN)   // 4096
// log2(10000) / 64  (for inv_freq = 10000^(-2j/64) = exp2(-(2j) * log2(1e4)/64))
#define ROPE_L2T_OVER_DK 0.20762050593046014f

__device__ __forceinline__ float red_max16(float v) {
  v = fmaxf(v, __shfl_xor(v, 1, 32));
  v = fmaxf(v, __shfl_xor(v, 2, 32));
  v = fmaxf(v, __shfl_xor(v, 4, 32));
  v = fmaxf(v, __shfl_xor(v, 8, 32));
  return v;
}
__device__ __forceinline__ float red_sum16(float v) {
  v += __shfl_xor(v, 1, 32);
  v += __shfl_xor(v, 2, 32);
  v += __shfl_xor(v, 4, 32);
  v += __shfl_xor(v, 8, 32);
  return v;
}

// C[m,n] = sum_k X[m,k] * W[n,k]   (M=4096, N=K=1024)
// Each wave computes a 32x64 macro-tile: 2 (M) x 4 (N) WMMA tiles.
// mode 0: plain store out[M,N]
// mode 1: RoPE + scatter to [B, H, S, DK]
// mode 2: scatter to [B, H, S, DK] (no RoPE)
__global__ __launch_bounds__(256) void gemm_wmma_f32(
    const float* __restrict__ X, const float* __restrict__ W,
    float* __restrict__ out, int mode)
{
  const int lane = threadIdx.x & 31;
  const int wave = threadIdx.x >> 5;
  const int half = lane >> 4;
  const int l16  = lane & 15;

  const int m0 = (blockIdx.y * 8 + wave) * 32;  // 32-row macro-tile base
  const int n0 = blockIdx.x * 64;               // 64-col macro-tile base

  const float* Arow0 = X + (size_t)(m0 + l16) * D_MODEL;        // rows m0..m0+15
  const float* Arow1 = Arow0 + (size_t)16 * D_MODEL;            // rows m0+16..m0+31
  const float* Brow0 = W + (size_t)(n0 + l16) * D_MODEL;        // cols n0+t*16
  const float* Brow1 = Brow0 + (size_t)16 * D_MODEL;
  const float* Brow2 = Brow0 + (size_t)32 * D_MODEL;
  const float* Brow3 = Brow0 + (size_t)48 * D_MODEL;

  v8f acc[2][4];
#pragma unroll
  for (int i = 0; i < 2; ++i)
#pragma unroll
    for (int t = 0; t < 4; ++t) acc[i][t] = (v8f){};

  const int koff = 2 * half;  // lanes 16-31 hold K+2, K+3
#pragma unroll 2
  for (int k = 0; k < D_MODEL; k += 4) {
    const int kk = k + koff;
    v2f a0 = *(const v2f*)(Arow0 + kk);
    v2f a1 = *(const v2f*)(Arow1 + kk);
    v2f b0 = *(const v2f*)(Brow0 + kk);
    v2f b1 = *(const v2f*)(Brow1 + kk);
    v2f b2 = *(const v2f*)(Brow2 + kk);
    v2f b3 = *(const v2f*)(Brow3 + kk);
    acc[0][0] = __builtin_amdgcn_wmma_f32_16x16x4_f32(false, a0, false, b0, (short)0, acc[0][0], false, false);
    acc[0][1] = __builtin_amdgcn_wmma_f32_16x16x4_f32(false, a0, false, b1, (short)0, acc[0][1], false, false);
    acc[0][2] = __builtin_amdgcn_wmma_f32_16x16x4_f32(false, a0, false, b2, (short)0, acc[0][2], false, false);
    acc[0][3] = __builtin_amdgcn_wmma_f32_16x16x4_f32(false, a0, false, b3, (short)0, acc[0][3], false, false);
    acc[1][0] = __builtin_amdgcn_wmma_f32_16x16x4_f32(false, a1, false, b0, (short)0, acc[1][0], false, false);
    acc[1][1] = __builtin_amdgcn_wmma_f32_16x16x4_f32(false, a1, false, b1, (short)0, acc[1][1], false, false);
    acc[1][2] = __builtin_amdgcn_wmma_f32_16x16x4_f32(false, a1, false, b2, (short)0, acc[1][2], false, false);
    acc[1][3] = __builtin_amdgcn_wmma_f32_16x16x4_f32(false, a1, false, b3, (short)0, acc[1][3], false, false);
  }

  if (mode == 0) {
#pragma unroll
    for (int i = 0; i < 2; ++i)
#pragma unroll
      for (int t = 0; t < 4; ++t)
#pragma unroll
        for (int r = 0; r < 8; ++r) {
          const int row = m0 + i * 16 + r + 8 * half;
          out[(size_t)row * D_MODEL + n0 + t * 16 + l16] = acc[i][t][r];
        }
  } else {
#pragma unroll
    for (int t = 0; t < 4; ++t) {
      const int gcol = n0 + t * 16 + l16;
      const int h = gcol >> 6;          // head
      const int d = gcol & 63;          // dim within head
      const bool even = (d & 1) == 0;
      const float invfreq = exp2f(-(float)(d & ~1) * ROPE_L2T_OVER_DK);
#pragma unroll
      for (int i = 0; i < 2; ++i)
#pragma unroll
        for (int r = 0; r < 8; ++r) {
          const int row = m0 + i * 16 + r + 8 * half;   // flat b*S + s
          const int b  = row >> 11;
          const int s  = row & (S_LEN - 1);
          float v  = acc[i][t][r];
          float vp = __shfl_xor(v, 1, 32);     // pair partner (col ^ 1)
          if (mode == 1) {
            const float ang = (float)s * invfreq;
            const float c = __cosf(ang), sn = __sinf(ang);
            // even: x*cos - x_pair*sin ; odd: x_pair*sin + x*cos
            v = v * c + (even ? -vp : vp) * sn;
          }
          out[(((size_t)(b * NHEAD + h) * S_LEN + s) * DK) + d] = v;
        }
    }
  }
}

// Flash attention, causal. One wave per (bh, 16-query tile).
// Q/K/V in [B,H,S,DK]; output merged-head [B,S,D].
__global__ __launch_bounds__(32) void flash_attn(
    const float* __restrict__ Q, const float* __restrict__ K,
    const float* __restrict__ V, float* __restrict__ att)
{
  __shared__ float p_lds[16 * 16];
  const int lane = threadIdx.x & 31;
  const int half = lane >> 4;
  const int l16  = lane & 15;
  const int qt   = blockIdx.x;            // 0..127 query tile
  const int bh   = blockIdx.y;            // 0..31  (b*16 + h)
  const size_t base = (size_t)bh * S_LEN * DK;

  // Preload Q A-frags for all 16 K-chunks, pre-scaled by 1/sqrt(dk)
  v2f qa[16];
  {
    const float* qrow = Q + base + (size_t)(qt * 16 + l16) * DK + 2 * half;
#pragma unroll
    for (int c = 0; c < 16; ++c) {
      v2f t = *(const v2f*)(qrow + 4 * c);
      qa[c] = t * 0.125f;
    }
  }

  v8f o0 = {}, o1 = {}, o2 = {}, o3 = {};
  float mrun[8], lrun[8];
#pragma unroll
  for (int r = 0; r < 8; ++r) { mrun[r] = -__builtin_inff(); lrun[r] = 0.f; }

  for (int kb = 0; kb <= qt; ++kb) {
    // ---- scores = Q K^T (16 WMMAs over dk=64) ----
    v8f sc = {};
    const float* krow = K + base + (size_t)(kb * 16 + l16) * DK + 2 * half;
#pragma unroll
    for (int c = 0; c < 16; ++c) {
      v2f bf = *(const v2f*)(krow + 4 * c);
      sc = __builtin_amdgcn_wmma_f32_16x16x4_f32(false, qa[c], false, bf,
                                                 (short)0, sc, false, false);
    }

    // ---- causal mask + streaming softmax ----
    float pvals[8], scl[8];
#pragma unroll
    for (int r = 0; r < 8; ++r) {
      const int row = r + 8 * half;
      const int qg  = qt * 16 + row;
      const int kg  = kb * 16 + l16;
      const float s = (kg <= qg) ? sc[r] : -__builtin_inff();
      const float rmax = red_max16(s);
      const float mnew = fmaxf(mrun[r], rmax);
      const float p = __expf(s - mnew);
      const float rsum = red_sum16(p);
      scl[r]  = __expf(mrun[r] - mnew);
      lrun[r] = lrun[r] * scl[r] + rsum;
      mrun[r] = mnew;
      pvals[r] = p;
    }

    // rescale accumulators (row mapping identical to score C-layout)
#pragma unroll
    for (int r = 0; r < 8; ++r) {
      o0[r] *= scl[r]; o1[r] *= scl[r]; o2[r] *= scl[r]; o3[r] *= scl[r];
    }

    // stage P through LDS: C-layout -> A-layout
#pragma unroll
    for (int r = 0; r < 8; ++r)
      p_lds[(r + 8 * half) * 16 + l16] = pvals[r];
    __syncthreads();

    // ---- out += P * V (4 kv-chunks x 4 n-tiles) ----
    const float* vbase = V + base + (size_t)(kb * 16) * DK;
#pragma unroll
    for (int c = 0; c < 4; ++c) {
      const int kk = c * 4 + 2 * half;
      v2f a;
      a[0] = p_lds[l16 * 16 + kk];
      a[1] = p_lds[l16 * 16 + kk + 1];
      v2f b0, b1, b2, b3;
      const float* v0 = vbase + (size_t)kk * DK + l16;
      const float* v1 = v0 + DK;
      b0[0] = v0[0];  b0[1] = v1[0];
      b1[0] = v0[16]; b1[1] = v1[16];
      b2[0] = v0[32]; b2[1] = v1[32];
      b3[0] = v0[48]; b3[1] = v1[48];
      o0 = __builtin_amdgcn_wmma_f32_16x16x4_f32(false, a, false, b0, (short)0, o0, false, false);
      o1 = __builtin_amdgcn_wmma_f32_16x16x4_f32(false, a, false, b1, (short)0, o1, false, false);
      o2 = __builtin_amdgcn_wmma_f32_16x16x4_f32(false, a, false, b2, (short)0, o2, false, false);
      o3 = __builtin_amdgcn_wmma_f32_16x16x4_f32(false, a, false, b3, (short)0, o3, false, false);
    }
    __syncthreads();
  }

  // ---- finalize: /= l, store merged-head [B,S,D] ----
  const int b = bh >> 4;
  const int h = bh & 15;
#pragma unroll
  for (int r = 0; r < 8; ++r) {
    const float inv = 1.0f / lrun[r];
    const int row = qt * 16 + r + 8 * half;
    const size_t ob = ((size_t)(b * S_LEN) + row) * D_MODEL + h * DK;
    att[ob +  0 + l16] = o0[r] * inv;
    att[ob + 16 + l16] = o1[r] * inv;
    att[ob + 32 + l16] = o2[r] * inv;
    att[ob + 48 + l16] = o3[r] * inv;
  }
}

extern "C" void kernel_launch(void* const* d_in, const int* in_sizes, int n_in,
                              void* d_out, int out_size, void* d_ws, size_t ws_size,
                              hipStream_t stream) {
  const float* x  = (const float*)d_in[0];
  const float* wq = (const float*)d_in[1];
  const float* wk = (const float*)d_in[2];
  const float* wv = (const float*)d_in[3];
  const float* wo = (const float*)d_in[4];
  float* out = (float*)d_out;

  float* ws  = (float*)d_ws;
  const size_t SZ = (size_t)M_TOT * D_MODEL;  // 4 M floats = 16 MB
  float* Qw  = ws;
  float* Kw  = ws + SZ;
  float* Vw  = ws + 2 * SZ;
  float* Aw  = ws + 3 * SZ;

  dim3 blk(256);
  // each block: 8 waves x (32x64) macro-tiles = 256 rows x 64 cols
  dim3 grd(D_MODEL / 64, M_TOT / (32 * 8));   // (16, 16)
  gemm_wmma_f32<<<grd, blk, 0, stream>>>(x, wq, Qw, 1);
  gemm_wmma_f32<<<grd, blk, 0, stream>>>(x, wk, Kw, 1);
  gemm_wmma_f32<<<grd, blk, 0, stream>>>(x, wv, Vw, 2);

  flash_attn<<<dim3(S_LEN / 16, BATCH * NHEAD), dim3(32), 0, stream>>>(Qw, Kw, Vw, Aw);

  gemm_wmma_f32<<<grd, blk, 0, stream>>>(Aw, wo, out, 0);
}